// GCN_1_12515534700681
// MI455X (gfx1250) — compile-verified
//
#include <hip/hip_runtime.h>

#define D 256
#define ALPHA 0.5f

typedef __attribute__((ext_vector_type(2))) float v2f;
typedef __attribute__((ext_vector_type(8))) float v8f;

// ---------------------------------------------------------------------------
// Kernel 1: support = input @ weight  via V_WMMA_F32_16X16X4_F32.
// One wave32 per 16x16 output tile. K loop: 256/4 = 64 WMMA issues.
//
// A 16x4 f32 fragment (2 VGPRs): lanes 0-15 -> M=lane,    v0=K, v1=K+1
//                                lanes 16-31 -> M=lane-16, v0=K+2, v1=K+3
// B 4x16 f32 fragment (2 VGPRs): mirrored (lane&15 = N column).
// C/D 16x16 f32 (8 VGPRs): VGPR j holds row M = j + 8*(lane>=16), N = lane&15.
// ---------------------------------------------------------------------------
__global__ void gcn_gemm_wmma(const float* __restrict__ A,   // [N, 256]
                              const float* __restrict__ W,   // [256, 256]
                              float* __restrict__ S,         // [N, 256]
                              int nTiles)                    // (N/16)*16
{
    const int lane    = threadIdx.x & 31;
    const int wave    = threadIdx.x >> 5;
    const int tile    = blockIdx.x * 8 + wave;
    if (tile >= nTiles) return;

    const int tileM   = tile >> 4;      // row-tile (N/16 of them)
    const int tileN   = tile & 15;      // col-tile (256/16 = 16 of them)
    const int half    = lane >> 4;      // 0: K,K+1 ; 1: K+2,K+3
    const int l15     = lane & 15;

    const int rowA    = tileM * 16 + l15;
    const int colB    = tileN * 16 + l15;
    const float* aRow = A + (size_t)rowA * D;

    v8f c = {};
#pragma unroll 4
    for (int k = 0; k < D; k += 4) {
        const int ka = k + half * 2;
        v2f a, b;
        a.x = aRow[ka];
        a.y = aRow[ka + 1];
        b.x = W[(size_t)ka * D + colB];
        b.y = W[(size_t)(ka + 1) * D + colB];
        // 8 args: (neg_a, A, neg_b, B, c_mod, C, reuse_a, reuse_b)
        c = __builtin_amdgcn_wmma_f32_16x16x4_f32(
                false, a, false, b, (short)0, c, false, false);
    }

#pragma unroll
    for (int j = 0; j < 8; ++j) {
        const int row = tileM * 16 + half * 8 + j;
        S[(size_t)row * D + colB] = c[j];
    }
}

// ---------------------------------------------------------------------------
// Kernel 2: zero the aggregation buffer (d_out doubles as agg accumulator).
// ---------------------------------------------------------------------------
__global__ void gcn_zero(float4* __restrict__ p, int n4)
{
    const int i = blockIdx.x * blockDim.x + threadIdx.x;
    if (i < n4) p[i] = make_float4(0.f, 0.f, 0.f, 0.f);
}

// ---------------------------------------------------------------------------
// Kernel 3: edge scatter: agg[dst] += w * support[src].
// One wave32 per edge; each lane owns 8 contiguous floats of the 256-row.
// support (51.2 MB) is L2-resident on MI455X (192 MB L2); atomics resolve
// in L2 (global_atomic_add_f32 via unsafeAtomicAdd).
// ---------------------------------------------------------------------------
__global__ void gcn_scatter(const float* __restrict__ S,
                            const long long* __restrict__ src,
                            const long long* __restrict__ dst,
                            const float* __restrict__ w,
                            float* __restrict__ agg,
                            int E)
{
    const int lane = threadIdx.x & 31;
    const int e    = (int)((blockIdx.x * (size_t)blockDim.x + threadIdx.x) >> 5);
    if (e >= E) return;

    const int   s  = (int)src[e];
    const int   d  = (int)dst[e];
    const float we = w[e];

    const float4* sp = (const float4*)(S + (size_t)s * D) + lane * 2;
    const float4 v0 = sp[0];
    const float4 v1 = sp[1];

    float* op = agg + (size_t)d * D + lane * 8;
    unsafeAtomicAdd(op + 0, we * v0.x);
    unsafeAtomicAdd(op + 1, we * v0.y);
    unsafeAtomicAdd(op + 2, we * v0.z);
    unsafeAtomicAdd(op + 3, we * v0.w);
    unsafeAtomicAdd(op + 4, we * v1.x);
    unsafeAtomicAdd(op + 5, we * v1.y);
    unsafeAtomicAdd(op + 6, we * v1.z);
    unsafeAtomicAdd(op + 7, we * v1.w);
}

// ---------------------------------------------------------------------------
// Kernel 4: out = relu(ALPHA*agg + (1-ALPHA)*init), in place on d_out.
// ---------------------------------------------------------------------------
__global__ void gcn_blend_relu(float4* __restrict__ out,
                               const float4* __restrict__ init,
                               int n4)
{
    const int i = blockIdx.x * blockDim.x + threadIdx.x;
    if (i >= n4) return;
    const float4 a = out[i];
    const float4 b = init[i];
    float4 r;
    r.x = fmaxf(0.f, fmaf(ALPHA, a.x, (1.f - ALPHA) * b.x));
    r.y = fmaxf(0.f, fmaf(ALPHA, a.y, (1.f - ALPHA) * b.y));
    r.z = fmaxf(0.f, fmaf(ALPHA, a.z, (1.f - ALPHA) * b.z));
    r.w = fmaxf(0.f, fmaf(ALPHA, a.w, (1.f - ALPHA) * b.w));
    out[i] = r;
}

extern "C" void kernel_launch(void* const* d_in, const int* in_sizes, int n_in,
                              void* d_out, int out_size, void* d_ws, size_t ws_size,
                              hipStream_t stream)
{
    const float*     input  = (const float*)d_in[0];      // [N, 256]
    const long long* adjSrc = (const long long*)d_in[1];  // [E]
    const long long* adjDst = (const long long*)d_in[2];  // [E]
    const float*     adjW   = (const float*)d_in[3];      // [E]
    const float*     init   = (const float*)d_in[4];      // [N, 256]
    const float*     weight = (const float*)d_in[5];      // [256, 256]
    float*           out    = (float*)d_out;              // [N, 256]
    float*           support = (float*)d_ws;              // [N, 256] scratch

    const int N = in_sizes[0] / D;   // 50000
    const int E = in_sizes[1];       // 800000

    // 1) support = input @ W   (WMMA f32 16x16x4)
    const int nTiles = (N / 16) * (D / 16);          // 50000 (N divisible by 16)
    const int gemmBlocks = (nTiles + 7) / 8;         // 8 waves/block
    gcn_gemm_wmma<<<gemmBlocks, 256, 0, stream>>>(input, weight, support, nTiles);

    // 2) zero agg (accumulate directly in d_out)
    const int n4 = (N * D) / 4;
    gcn_zero<<<(n4 + 255) / 256, 256, 0, stream>>>((float4*)out, n4);

    // 3) scatter-add edges: one wave per edge
    const long long scatterThreads = (long long)E * 32;
    const int scatterBlocks = (int)((scatterThreads + 255) / 256);
    gcn_scatter<<<scatterBlocks, 256, 0, stream>>>(support, adjSrc, adjDst, adjW, out, E);

    // 4) blend + ReLU in place
    gcn_blend_relu<<<(n4 + 255) / 256, 256, 0, stream>>>((float4*)out, (const float4*)init, n4);
}